// ToposTransformer_36833639530796
// MI455X (gfx1250) — compile-verified
//
#include <hip/hip_runtime.h>
#include <hip/hip_bf16.h>
#include <math.h>

// ---------------- problem constants ----------------
#define V_    8192
#define D_    128
#define H_    8
#define L_    2
#define S_    128
#define HD_   16
#define HID_  341
#define HIDP_ 352   // 22 * 16, zero-padded
#define EPS_  1e-6f

typedef __attribute__((ext_vector_type(2))) float v2f;
typedef __attribute__((ext_vector_type(8))) float v8f;

__device__ __forceinline__ float sigm_(float x) { return 1.0f / (1.0f + __expf(-x)); }

// ---------------- WMMA f32 GEMM: C[tile] (+)= A @ B ----------------
// One wave (32 threads) per 16x16 output tile. Batched via blockIdx.y.
// A: MxK row-major (lda), B: KxN row-major (ldb), C: MxN row-major (ldc).
// K must be a multiple of 4. Uses V_WMMA_F32_16X16X4_F32 (full fp32 precision).
__global__ void gemm_nn_wmma(const float* __restrict__ A, int lda, long long Abs,
                             const float* __restrict__ B, int ldb, long long Bbs,
                             float* __restrict__ C, int ldc, long long Cbs,
                             int K, int Ntiles, int acc) {
    const int lane = threadIdx.x;         // 0..31, EXEC all ones
    const int half = lane >> 4;           // 0/1
    const int lo   = lane & 15;
    const int tile = blockIdx.x;
    const int m0 = (tile / Ntiles) << 4;
    const int n0 = (tile % Ntiles) << 4;
    const long long b = blockIdx.y;
    A += b * Abs; B += b * Bbs; C += b * Cbs;

    v8f c;
    if (acc) {
        #pragma unroll
        for (int r = 0; r < 8; ++r)
            c[r] = C[(long long)(m0 + r + 8 * half) * ldc + n0 + lo];
    } else {
        #pragma unroll
        for (int r = 0; r < 8; ++r) c[r] = 0.0f;
    }
    for (int k0 = 0; k0 < K; k0 += 4) {
        v2f a, bb;
        // A 16x4 fragment: lane = M, half-wave selects K pair {0,1} / {2,3}
        a.x = A[(long long)(m0 + lo) * lda + k0 + 2 * half];
        a.y = A[(long long)(m0 + lo) * lda + k0 + 2 * half + 1];
        // B 4x16 fragment: lane = N, half-wave selects K pair
        bb.x = B[(long long)(k0 + 2 * half)     * ldb + n0 + lo];
        bb.y = B[(long long)(k0 + 2 * half + 1) * ldb + n0 + lo];
        c = __builtin_amdgcn_wmma_f32_16x16x4_f32(false, a, false, bb,
                                                  (short)0, c, false, false);
    }
    #pragma unroll
    for (int r = 0; r < 8; ++r)
        C[(long long)(m0 + r + 8 * half) * ldc + n0 + lo] = c[r];
}

// ---------------- attention scores: Sc[h,i,j] = scale * q_h[i]·k_h[j], causal mask fused ----------------
// grid.x = 64 tiles (8x8), grid.y = head. A = Q (+h*16), B = K^T (NT form).
__global__ void attn_scores_wmma(const float* __restrict__ Q,
                                 const float* __restrict__ Km,
                                 float* __restrict__ Sc, float scale) {
    const int lane = threadIdx.x;
    const int half = lane >> 4;
    const int lo   = lane & 15;
    const int m0 = (blockIdx.x >> 3) << 4;
    const int n0 = (blockIdx.x & 7) << 4;
    const int h  = blockIdx.y;
    const float* Ah = Q  + h * HD_;
    const float* Bh = Km + h * HD_;

    v8f c;
    #pragma unroll
    for (int r = 0; r < 8; ++r) c[r] = 0.0f;
    #pragma unroll
    for (int k0 = 0; k0 < HD_; k0 += 4) {
        v2f a, bb;
        a.x  = Ah[(m0 + lo) * D_ + k0 + 2 * half];
        a.y  = Ah[(m0 + lo) * D_ + k0 + 2 * half + 1];
        bb.x = Bh[(n0 + lo) * D_ + k0 + 2 * half];       // B[k][n] = K[n][k]
        bb.y = Bh[(n0 + lo) * D_ + k0 + 2 * half + 1];
        c = __builtin_amdgcn_wmma_f32_16x16x4_f32(false, a, false, bb,
                                                  (short)0, c, false, false);
    }
    float* out = Sc + (long long)h * S_ * S_;
    #pragma unroll
    for (int r = 0; r < 8; ++r) {
        int i = m0 + r + 8 * half;
        int j = n0 + lo;
        out[i * S_ + j] = (j <= i) ? c[r] * scale : -1.0e9f;
    }
}

// ---------------- row softmax over S=128 (grid = H*S rows, 128 thr) ----------------
__global__ void softmax_rows(float* __restrict__ Sc) {
    __shared__ float red[128];
    const int t = threadIdx.x;
    float* p = Sc + (long long)blockIdx.x * S_;
    float v = p[t];
    red[t] = v; __syncthreads();
    for (int k = 64; k > 0; k >>= 1) { if (t < k) red[t] = fmaxf(red[t], red[t + k]); __syncthreads(); }
    float m = red[0]; __syncthreads();
    float e = __expf(v - m);
    red[t] = e; __syncthreads();
    for (int k = 64; k > 0; k >>= 1) { if (t < k) red[t] += red[t + k]; __syncthreads(); }
    p[t] = e / red[0];
}

// ---------------- rmsnorm (+ optional fused L2 row normalize) ----------------
__global__ void rmsnorm_kernel(const float* __restrict__ X, const float* __restrict__ W,
                               float* __restrict__ Y, int l2norm) {
    __shared__ float red[128];
    const int s = blockIdx.x, t = threadIdx.x;
    float x = X[s * D_ + t];
    red[t] = x * x; __syncthreads();
    for (int k = 64; k > 0; k >>= 1) { if (t < k) red[t] += red[t + k]; __syncthreads(); }
    float ms = red[0] * (1.0f / D_); __syncthreads();
    float y = x * rsqrtf(ms + EPS_) * W[t];
    if (l2norm) {
        red[t] = y * y; __syncthreads();
        for (int k = 64; k > 0; k >>= 1) { if (t < k) red[t] += red[t + k]; __syncthreads(); }
        y *= rsqrtf(red[0]);
    }
    Y[s * D_ + t] = y;
}

// ---------------- RoPE on q and k in place (grid = S, 64 thr: h*8+p) ----------------
__global__ void rope_kernel(float* __restrict__ Q, float* __restrict__ Km) {
    const int s = blockIdx.x;
    const int h = threadIdx.x >> 3, p = threadIdx.x & 7;
    float ang = (float)s * __powf(10000.0f, -(float)p / 8.0f);
    float cs = __cosf(ang), sn = __sinf(ang);
    int base = s * D_ + h * HD_ + 2 * p;
    float q0 = Q[base], q1 = Q[base + 1];
    Q[base] = q0 * cs - q1 * sn; Q[base + 1] = q0 * sn + q1 * cs;
    float k0 = Km[base], k1 = Km[base + 1];
    Km[base] = k0 * cs - k1 * sn; Km[base + 1] = k0 * sn + k1 * cs;
}

// ---------------- embedding gather + sigmoid ----------------
__global__ void embed_kernel(const int* __restrict__ idx, const float* __restrict__ w_raw,
                             float* __restrict__ X) {
    const int s = blockIdx.x, d = threadIdx.x;
    X[s * D_ + d] = sigm_(w_raw[(long long)d * V_ + idx[s]]);
}

// ---------------- v_n[v,d] = sigmoid(w_raw[d,v]) / ||sigmoid(w_raw[:,v])|| ----------------
__global__ void vn_kernel(const float* __restrict__ w_raw, float* __restrict__ VN) {
    __shared__ float red[128];
    const int v = blockIdx.x, d = threadIdx.x;
    float t = sigm_(w_raw[(long long)d * V_ + v]);
    red[d] = t * t; __syncthreads();
    for (int k = 64; k > 0; k >>= 1) { if (d < k) red[d] += red[d + k]; __syncthreads(); }
    VN[(long long)v * D_ + d] = t * rsqrtf(red[0]);
}

// ---------------- zero-pad copy (for HID 341 -> 352) ----------------
__global__ void pad_kernel(const float* __restrict__ src, int inR, int inC,
                           float* __restrict__ dst, int outC, int total) {
    int i = blockIdx.x * blockDim.x + threadIdx.x;
    if (i >= total) return;
    int r = i / outC, c = i % outC;
    dst[i] = (r < inR && c < inC) ? src[r * inC + c] : 0.0f;
}

// ---------------- A1 = silu(A1) * A3 ----------------
__global__ void silu_mul_kernel(float* __restrict__ A1, const float* __restrict__ A3, int n) {
    int i = blockIdx.x * blockDim.x + threadIdx.x;
    if (i < n) {
        float a = A1[i];
        A1[i] = (a / (1.0f + __expf(-a))) * A3[i];
    }
}

// ---------------- logits: mean_d( xn[s,d] <= vn[v,d] ? 1 : vn[v,d] ), clipped ----------------
__global__ void logits_kernel(const float* __restrict__ XN, const float* __restrict__ VN,
                              float* __restrict__ OUT) {
    __shared__ float xs[D_];
    const int s = blockIdx.y, t = threadIdx.x;
    if (t < D_) xs[t] = XN[s * D_ + t];
    __syncthreads();
    const int v = blockIdx.x * 256 + t;
    const float* vp = VN + (long long)v * D_;
    float acc = 0.0f;
    #pragma unroll 4
    for (int d = 0; d < D_; ++d) {
        float b = vp[d];
        acc += (xs[d] <= b) ? 1.0f : b;
    }
    float lg = acc * (1.0f / D_);
    OUT[(long long)s * V_ + v] = fminf(fmaxf(lg, 1e-6f), 1.0f - 1e-6f);
}

// ---------------- host orchestration ----------------
extern "C" void kernel_launch(void* const* d_in, const int* in_sizes, int n_in,
                              void* d_out, int out_size, void* d_ws, size_t ws_size,
                              hipStream_t stream) {
    (void)in_sizes; (void)n_in; (void)out_size; (void)ws_size;
    const int*   idx   = (const int*)d_in[0];
    const float* w_raw = (const float*)d_in[1];
    const float* n1w   = (const float*)d_in[2];
    const float* n2w   = (const float*)d_in[3];
    const float* wq    = (const float*)d_in[4];
    const float* wk    = (const float*)d_in[5];
    const float* wv    = (const float*)d_in[6];
    const float* wo    = (const float*)d_in[7];
    const float* w1    = (const float*)d_in[8];   // note setup_inputs order: w1, w3, w2
    const float* w3    = (const float*)d_in[9];
    const float* w2    = (const float*)d_in[10];
    const float* fnw   = (const float*)d_in[11];
    float* out = (float*)d_out;

    float* W   = (float*)d_ws;
    float* x   = W;                 // 128*128
    float* h   = x   + 16384;
    float* q   = h   + 16384;
    float* k   = q   + 16384;
    float* v   = k   + 16384;
    float* o   = v   + 16384;
    float* sc  = o   + 16384;       // H*S*S = 131072
    float* w1p = sc  + 131072;      // 128*352
    float* w3p = w1p + 45056;
    float* w2p = w3p + 45056;       // 352*128
    float* a1  = w2p + 45056;
    float* a3  = a1  + 45056;
    float* vn  = a3  + 45056;       // 8192*128
    float* xn  = vn  + 1048576;     // 128*128

    vn_kernel<<<V_, 128, 0, stream>>>(w_raw, vn);
    embed_kernel<<<S_, 128, 0, stream>>>(idx, w_raw, x);

    const float scale = 1.0f / sqrtf((float)HD_);
    const int tot1 = D_ * HIDP_;    // 45056
    const int tot2 = HIDP_ * D_;

    for (int i = 0; i < L_; ++i) {
        // ---- attention ----
        rmsnorm_kernel<<<S_, 128, 0, stream>>>(x, n1w + i * D_, h, 0);
        gemm_nn_wmma<<<dim3(64, 1), 32, 0, stream>>>(h, D_, 0, wq + i * D_ * D_, D_, 0,
                                                     q, D_, 0, D_, 8, 0);
        gemm_nn_wmma<<<dim3(64, 1), 32, 0, stream>>>(h, D_, 0, wk + i * D_ * D_, D_, 0,
                                                     k, D_, 0, D_, 8, 0);
        gemm_nn_wmma<<<dim3(64, 1), 32, 0, stream>>>(h, D_, 0, wv + i * D_ * D_, D_, 0,
                                                     v, D_, 0, D_, 8, 0);
        rope_kernel<<<S_, 64, 0, stream>>>(q, k);
        attn_scores_wmma<<<dim3(64, H_), 32, 0, stream>>>(q, k, sc, scale);
        softmax_rows<<<H_ * S_, 128, 0, stream>>>(sc);
        // o[:, h*16:(h+1)*16] = attn_h @ v_h   (batched over heads)
        gemm_nn_wmma<<<dim3(8, H_), 32, 0, stream>>>(sc, S_, (long long)S_ * S_,
                                                     v, D_, 16,
                                                     o, D_, 16, S_, 1, 0);
        // x += o @ wo
        gemm_nn_wmma<<<dim3(64, 1), 32, 0, stream>>>(o, D_, 0, wo + i * D_ * D_, D_, 0,
                                                     x, D_, 0, D_, 8, 1);
        // ---- FFN ----
        rmsnorm_kernel<<<S_, 128, 0, stream>>>(x, n2w + i * D_, h, 0);
        pad_kernel<<<(tot1 + 255) / 256, 256, 0, stream>>>(w1 + i * D_ * HID_, D_, HID_, w1p, HIDP_, tot1);
        pad_kernel<<<(tot1 + 255) / 256, 256, 0, stream>>>(w3 + i * D_ * HID_, D_, HID_, w3p, HIDP_, tot1);
        pad_kernel<<<(tot2 + 255) / 256, 256, 0, stream>>>(w2 + i * HID_ * D_, HID_, D_, w2p, D_, tot2);
        gemm_nn_wmma<<<dim3(8 * 22, 1), 32, 0, stream>>>(h, D_, 0, w1p, HIDP_, 0,
                                                         a1, HIDP_, 0, D_, 22, 0);
        gemm_nn_wmma<<<dim3(8 * 22, 1), 32, 0, stream>>>(h, D_, 0, w3p, HIDP_, 0,
                                                         a3, HIDP_, 0, D_, 22, 0);
        silu_mul_kernel<<<(tot1 + 255) / 256, 256, 0, stream>>>(a1, a3, tot1);
        // x += g @ w2   (pad rows of w2p are zero -> exact)
        gemm_nn_wmma<<<dim3(64, 1), 32, 0, stream>>>(a1, HIDP_, 0, w2p, D_, 0,
                                                     x, D_, 0, HIDP_, 8, 1);
    }

    // final rmsnorm + L2 row normalize, then compare-select-mean logits
    rmsnorm_kernel<<<S_, 128, 0, stream>>>(x, fnw, xn, 1);
    logits_kernel<<<dim3(V_ / 256, S_), 256, 0, stream>>>(xn, vn, out);
}